// CurvatureStdLoss_46557445488915
// MI455X (gfx1250) — compile-verified
//
#include <hip/hip_runtime.h>
#include <hip/hip_bf16.h>
#include <math.h>

// CurvatureStdLoss for MI455X (gfx1250, wave32).
// Distance cross-terms computed with V_WMMA_F32_16X16X4_F32 (K=3 padded to 4).

typedef __attribute__((ext_vector_type(2))) float v2f;
typedef __attribute__((ext_vector_type(8))) float v8f;

#define KNN 16          // reference k
#define TOPK (KNN + 1)  // k+1, slot 0 = self

// ---------------------------------------------------------------------------
// kappa kernel: for each query row, find 17 nearest neighbors within pc
// (self forced to slot 0), then kappa = mean_j |dot(normalize(nn_j - p), n_p)|
// One wave32 per 16-row query tile.
// ---------------------------------------------------------------------------
__global__ void __launch_bounds__(32)
kappa_kernel(const float* __restrict__ pc,
             const float* __restrict__ normals,
             float* __restrict__ kappa,
             int N) {
    __shared__ float sh_qn[16];
    __shared__ float sh_dist[16][17];  // padded to dodge bank conflicts

    const int ntiles = N >> 4;
    const int b      = blockIdx.x / ntiles;
    const int qtile  = blockIdx.x % ntiles;
    const int qbase  = qtile << 4;

    const int lane = threadIdx.x;      // 0..31
    const int half = lane >> 4;        // 0: K={0,1}, 1: K={2,3(pad)}
    const int lid  = lane & 15;

    const float* pcb = pc      + (size_t)b * N * 3;
    const float* nrb = normals + (size_t)b * N * 3;

    // --- load this lane's query point (row m = lid) -------------------------
    const int qrow = qbase + lid;
    float qx = pcb[qrow * 3 + 0];
    float qy = pcb[qrow * 3 + 1];
    float qz = pcb[qrow * 3 + 2];
    float qn = qx * qx + qy * qy + qz * qz;
    if (lane < 16) sh_qn[lid] = qn;
    __syncthreads();

    // A operand (16x4 f32): lanes 0-15 -> K=0,1 ; lanes 16-31 -> K=2,3
    v2f Aop;
    Aop.x = half ? qz : qx;
    Aop.y = half ? 0.0f : qy;

    // Row norms this lane needs for its 8 accumulator rows (M = v + half*8)
    float rn[8];
#pragma unroll
    for (int v = 0; v < 8; ++v) rn[v] = sh_qn[v + half * 8];

    // --- top-17 state (lanes 0-15), fully in registers ----------------------
    float bd[TOPK];
    int   bi[TOPK];
#pragma unroll
    for (int j = 0; j < TOPK; ++j) { bd[j] = 3.0e38f; bi[j] = 0; }

    // --- sweep candidate tiles ---------------------------------------------
    for (int ct = 0; ct < ntiles; ++ct) {
        const int cbase = ct << 4;
        const int crow  = cbase + lid;
        float cx = pcb[crow * 3 + 0];
        float cy = pcb[crow * 3 + 1];
        float cz = pcb[crow * 3 + 2];
        float cn = cx * cx + cy * cy + cz * cz;

        // B operand (4x16 f32): lanes 0-15 -> K=0,1 ; lanes 16-31 -> K=2,3
        v2f Bop;
        Bop.x = half ? cz : cx;
        Bop.y = half ? 0.0f : cy;

        v8f acc = {};
        acc = __builtin_amdgcn_wmma_f32_16x16x4_f32(
            /*neg_a=*/false, Aop, /*neg_b=*/false, Bop,
            /*c_mod=*/(short)0, acc, /*reuse_a=*/false, /*reuse_b=*/false);

        // d[m][n] = |q_m|^2 + |c_n|^2 - 2 q_m.c_n ; lane owns column n = lid,
        // VGPR v holds row m = v + half*8
#pragma unroll
        for (int v = 0; v < 8; ++v) {
            const int m = v + half * 8;
            float d = rn[v] + cn - 2.0f * acc[v];
            if (qbase + m == cbase + lid) d = -3.0e38f;  // self -> slot 0
            sh_dist[m][lid] = d;
        }
        __syncthreads();

        // lanes 0-15: scan 16 columns of own row, guarded insertion
        if (lane < 16) {
#pragma unroll
            for (int cc = 0; cc < 16; ++cc) {
                float d = sh_dist[lid][cc];
                if (d < bd[TOPK - 1]) {
                    float cd = d;
                    int   ci = cbase + cc;
#pragma unroll
                    for (int p = 0; p < TOPK; ++p) {
                        if (cd < bd[p]) {
                            float td = bd[p]; bd[p] = cd; cd = td;
                            int   ti = bi[p]; bi[p] = ci; ci = ti;
                        }
                    }
                }
            }
        }
        __syncthreads();
    }

    // --- kappa from neighbors 1..16 (slot 0 is self) ------------------------
    if (lane < 16) {
        float nx = nrb[qrow * 3 + 0];
        float ny = nrb[qrow * 3 + 1];
        float nz = nrb[qrow * 3 + 2];
        float s = 0.0f;
#pragma unroll
        for (int j = 1; j <= KNN; ++j) {
            const int ni = bi[j];
            float vx = pcb[ni * 3 + 0] - qx;
            float vy = pcb[ni * 3 + 1] - qy;
            float vz = pcb[ni * 3 + 2] - qz;
            float len = sqrtf(vx * vx + vy * vy + vz * vz) + 1e-12f;
            float inv = 1.0f / len;
            s += fabsf((vx * nx + vy * ny + vz * nz) * inv);
        }
        kappa[(size_t)b * N + qrow] = s * (1.0f / (float)KNN);
    }
}

// ---------------------------------------------------------------------------
// nn_normals kernel: adv_normals[i] = ori_normals[argmin_j ||adv_i - ori_j||^2]
// Same WMMA tiling; per-row running argmin (first index wins ties).
// ---------------------------------------------------------------------------
__global__ void __launch_bounds__(32)
nn_normals_kernel(const float* __restrict__ adv,
                  const float* __restrict__ ori,
                  const float* __restrict__ ori_nrm,
                  float* __restrict__ adv_nrm,
                  int N) {
    __shared__ float sh_qn[16];
    __shared__ float sh_dist[16][17];

    const int ntiles = N >> 4;
    const int b      = blockIdx.x / ntiles;
    const int qtile  = blockIdx.x % ntiles;
    const int qbase  = qtile << 4;

    const int lane = threadIdx.x;
    const int half = lane >> 4;
    const int lid  = lane & 15;

    const float* ab = adv     + (size_t)b * N * 3;
    const float* ob = ori     + (size_t)b * N * 3;
    const float* nb = ori_nrm + (size_t)b * N * 3;

    const int qrow = qbase + lid;
    float qx = ab[qrow * 3 + 0];
    float qy = ab[qrow * 3 + 1];
    float qz = ab[qrow * 3 + 2];
    if (lane < 16) sh_qn[lid] = qx * qx + qy * qy + qz * qz;
    __syncthreads();

    v2f Aop;
    Aop.x = half ? qz : qx;
    Aop.y = half ? 0.0f : qy;

    float rn[8];
#pragma unroll
    for (int v = 0; v < 8; ++v) rn[v] = sh_qn[v + half * 8];

    float best_d = 3.0e38f;
    int   best_i = 0;

    for (int ct = 0; ct < ntiles; ++ct) {
        const int cbase = ct << 4;
        const int crow  = cbase + lid;
        float cx = ob[crow * 3 + 0];
        float cy = ob[crow * 3 + 1];
        float cz = ob[crow * 3 + 2];
        float cn = cx * cx + cy * cy + cz * cz;

        v2f Bop;
        Bop.x = half ? cz : cx;
        Bop.y = half ? 0.0f : cy;

        v8f acc = {};
        acc = __builtin_amdgcn_wmma_f32_16x16x4_f32(
            false, Aop, false, Bop, (short)0, acc, false, false);

#pragma unroll
        for (int v = 0; v < 8; ++v) {
            const int m = v + half * 8;
            sh_dist[m][lid] = rn[v] + cn - 2.0f * acc[v];
        }
        __syncthreads();

        if (lane < 16) {
#pragma unroll
            for (int cc = 0; cc < 16; ++cc) {
                float d = sh_dist[lid][cc];
                if (d < best_d) { best_d = d; best_i = cbase + cc; }
            }
        }
        __syncthreads();
    }

    if (lane < 16) {
        adv_nrm[(size_t)qrow * 3 + 0 + (size_t)b * N * 3] = nb[best_i * 3 + 0];
        adv_nrm[(size_t)qrow * 3 + 1 + (size_t)b * N * 3] = nb[best_i * 3 + 1];
        adv_nrm[(size_t)qrow * 3 + 2 + (size_t)b * N * 3] = nb[best_i * 3 + 2];
    }
}

// ---------------------------------------------------------------------------
// Final reduction: std(kappa, ddof=1) per batch for ori & adv, mean |diff|.
// ---------------------------------------------------------------------------
__global__ void __launch_bounds__(256)
std_loss_kernel(const float* __restrict__ ko,
                const float* __restrict__ ka,
                float* __restrict__ out,
                int B, int N) {
    __shared__ float red[256];
    const int tid = threadIdx.x;
    float loss = 0.0f;

    for (int b = 0; b < B; ++b) {
        const float* kob = ko + (size_t)b * N;
        const float* kab = ka + (size_t)b * N;
        float s1o = 0.0f, s2o = 0.0f, s1a = 0.0f, s2a = 0.0f;
        for (int i = tid; i < N; i += 256) {
            float vo = kob[i], va = kab[i];
            s1o += vo; s2o += vo * vo;
            s1a += va; s2a += va * va;
        }
        float vals[4] = {s1o, s2o, s1a, s2a};
        float r[4];
#pragma unroll
        for (int k2 = 0; k2 < 4; ++k2) {
            red[tid] = vals[k2];
            __syncthreads();
            for (int off = 128; off > 0; off >>= 1) {
                if (tid < off) red[tid] += red[tid + off];
                __syncthreads();
            }
            r[k2] = red[0];
            __syncthreads();
        }
        const float fn = (float)N;
        float varo = (r[1] - r[0] * r[0] / fn) / (fn - 1.0f);
        float vara = (r[3] - r[2] * r[2] / fn) / (fn - 1.0f);
        float so = sqrtf(fmaxf(varo, 0.0f));
        float sa = sqrtf(fmaxf(vara, 0.0f));
        loss += fabsf(sa - so);
    }
    if (tid == 0) out[0] = loss / (float)B;
}

// ---------------------------------------------------------------------------
extern "C" void kernel_launch(void* const* d_in, const int* in_sizes, int n_in,
                              void* d_out, int out_size, void* d_ws, size_t ws_size,
                              hipStream_t stream) {
    const float* ori = (const float*)d_in[0];   // (B,N,3) f32
    const float* adv = (const float*)d_in[1];   // (B,N,3) f32
    const float* nrm = (const float*)d_in[2];   // (B,N,3) f32
    // d_in[3] is k (==16), baked into KNN at compile time per reference.

    const int B = 4;                      // reference constant
    const int N = in_sizes[0] / (B * 3);  // 4096

    float* ws      = (float*)d_ws;
    float* k_ori   = ws;                          // B*N
    float* k_adv   = ws + (size_t)B * N;          // B*N
    float* adv_nrm = ws + (size_t)2 * B * N;      // B*N*3

    const int ntiles = N / 16;
    dim3 grid(B * ntiles), block(32);

    kappa_kernel<<<grid, block, 0, stream>>>(ori, nrm, k_ori, N);
    nn_normals_kernel<<<grid, block, 0, stream>>>(adv, ori, nrm, adv_nrm, N);
    kappa_kernel<<<grid, block, 0, stream>>>(adv, adv_nrm, k_adv, N);
    std_loss_kernel<<<1, 256, 0, stream>>>(k_ori, k_adv, (float*)d_out, B, N);
}